// NLUTNet_12979391168851
// MI455X (gfx1250) — compile-verified
//
#include <hip/hip_runtime.h>
#include <hip/hip_bf16.h>

typedef __attribute__((ext_vector_type(2))) float v2f;
typedef __attribute__((ext_vector_type(8))) float v8f;

#define DIM   33
#define DD    1089      // 33*33
#define DDD   35937     // 33^3
#define NUM   20
#define SL    5
#define WL    20
#define NBAT  8
#define HW    (1024*1024)

// ---------------------------------------------------------------------------
// Stage 1: T = luts(300x20) @ w_layers(20x1089), f32 WMMA 16x16x4.
// One wave (32 threads) computes a 16x16 tile of T. K=20 -> 5 chunks of 4.
// A-frag (16x4 f32): lane m=L%16, VGPR0 K=khalf, VGPR1 K=khalf+1 where
// khalf = (L>=16 ? 2 : 0). B-frag (4x16 f32) mirrors: VGPR v holds row
// K = khalf + v for column n = L%16. C/D: VGPR v -> row v + (L>=16 ? 8:0).
// ---------------------------------------------------------------------------
__global__ __launch_bounds__(32)
void nlut_gemm_wmma(const float* __restrict__ A,   // luts   (300 x 20)
                    const float* __restrict__ B,   // w_lay  (20 x 1089)
                    float* __restrict__ T)         // out    (300 x 1089)
{
    const int M = SL * NUM * 3;   // 300
    const int K = WL;             // 20
    const int N = DD;             // 1089

    const int tileM = blockIdx.x;        // 0..18
    const int tileN = blockIdx.y;        // 0..68
    const int lane  = threadIdx.x;       // 0..31

    const int m      = tileM * 16 + (lane & 15);
    const int n      = tileN * 16 + (lane & 15);
    const int khalf  = (lane >> 4) * 2;  // 0 or 2

    v8f acc = {};
    #pragma unroll
    for (int k0 = 0; k0 < K; k0 += 4) {
        const int ka = k0 + khalf;
        v2f a, b;
        a.x = (m < M) ? A[m * K + ka]         : 0.0f;
        a.y = (m < M) ? A[m * K + ka + 1]     : 0.0f;
        b.x = (n < N) ? B[ka * N + n]         : 0.0f;
        b.y = (n < N) ? B[(ka + 1) * N + n]   : 0.0f;
        acc = __builtin_amdgcn_wmma_f32_16x16x4_f32(
                  /*neg_a=*/false, a, /*neg_b=*/false, b,
                  /*c_mod=*/(short)0, acc, /*reuse_a=*/false, /*reuse_b=*/false);
    }

    const int rbase = tileM * 16 + ((lane >> 4) * 8);
    if (n < N) {
        #pragma unroll
        for (int v = 0; v < 8; ++v) {
            const int mm = rbase + v;
            if (mm < M) T[mm * N + n] = acc[v];
        }
    }
}

// ---------------------------------------------------------------------------
// Stage 2: V[b][ch][s][k] = sum_n pw[b,n] * T[s*60 + n*3 + ch, k]
// ---------------------------------------------------------------------------
__global__ __launch_bounds__(256)
void nlut_fuse_pw(const float* __restrict__ T,
                  const float* __restrict__ pw,
                  float* __restrict__ V)
{
    const int idx   = blockIdx.x * blockDim.x + threadIdx.x;
    const int total = NBAT * 3 * SL * DD;      // 130680
    if (idx >= total) return;

    const int k  = idx % DD;
    const int s  = (idx / DD) % SL;
    const int ch = (idx / (DD * SL)) % 3;
    const int b  = idx / (DD * SL * 3);

    float acc = 0.0f;
    #pragma unroll
    for (int n = 0; n < NUM; ++n)
        acc += pw[b * NUM + n] * T[(s * (NUM * 3) + n * 3 + ch) * DD + k];
    V[idx] = acc;
}

// ---------------------------------------------------------------------------
// Stage 3: fused LUT with cube_to_lut permutation.
// lut[b][ch][z][y][x]:
//   ch==0: (a,i,j) = (x, z, y)
//   ch==1: (a,i,j) = (y, z, x)
//   ch==2: (a,i,j) = (z, y, x)
// lut = sum_s s_layers[a,s] * V[b][ch][s][i*33+j]
// ---------------------------------------------------------------------------
__global__ __launch_bounds__(256)
void nlut_build_lut(const float* __restrict__ V,
                    const float* __restrict__ sl,   // (33 x 5)
                    float* __restrict__ lut)
{
    const int idx   = blockIdx.x * blockDim.x + threadIdx.x;
    const int total = NBAT * 3 * DDD;          // 862488
    if (idx >= total) return;

    const int x  = idx % DIM;
    const int y  = (idx / DIM) % DIM;
    const int z  = (idx / DD) % DIM;
    const int ch = (idx / DDD) % 3;
    const int b  = idx / (DDD * 3);

    int a, k;
    if (ch == 0)      { a = x; k = z * DIM + y; }
    else if (ch == 1) { a = y; k = z * DIM + x; }
    else              { a = z; k = y * DIM + x; }

    const float* Vb = V + ((size_t)(b * 3 + ch) * SL) * DD;
    float acc = 0.0f;
    #pragma unroll
    for (int s = 0; s < SL; ++s)
        acc += sl[a * SL + s] * Vb[s * DD + k];
    lut[idx] = acc;
}

// ---------------------------------------------------------------------------
// Stage 4: trilinear apply + residual. Memory bound (~201 MB @ 23.3 TB/s).
// One thread handles 4 consecutive pixels (float4 streaming loads/stores);
// the 3.4 MB fused LUT stays L2-resident, 24 scattered b32 gathers/pixel.
// ---------------------------------------------------------------------------
__global__ __launch_bounds__(256)
void nlut_apply(const float* __restrict__ img,
                const float* __restrict__ lut,
                float* __restrict__ out)
{
    const int idx = blockIdx.x * blockDim.x + threadIdx.x;   // quad id
    const int b   = idx >> 18;                 // HW/4 = 2^18 quads per image
    const int p4  = (idx & ((HW / 4) - 1)) * 4;

    const float scale = 32.0f / 1.000001f;     // 1/binsize

    const size_t ib = (size_t)b * 3 * HW + p4;
    const float4 r4 = *(const float4*)(img + ib);
    const float4 g4 = *(const float4*)(img + ib + HW);
    const float4 b4 = *(const float4*)(img + ib + 2 * (size_t)HW);

    const float* L0 = lut + (size_t)(b * 3 + 0) * DDD;
    const float* L1 = lut + (size_t)(b * 3 + 1) * DDD;
    const float* L2 = lut + (size_t)(b * 3 + 2) * DDD;

    float4 o0, o1, o2;
    float* o0p = (float*)&o0; float* o1p = (float*)&o1; float* o2p = (float*)&o2;
    const float* rp = (const float*)&r4;
    const float* gp = (const float*)&g4;
    const float* bp = (const float*)&b4;

    #pragma unroll
    for (int j = 0; j < 4; ++j) {
        const float r = rp[j], g = gp[j], bl = bp[j];
        const float xr = r * scale, xg = g * scale, xb = bl * scale;
        int ri = (int)floorf(xr); ri = ri < 0 ? 0 : (ri > 31 ? 31 : ri);
        int gi = (int)floorf(xg); gi = gi < 0 ? 0 : (gi > 31 ? 31 : gi);
        int bi = (int)floorf(xb); bi = bi < 0 ? 0 : (bi > 31 ? 31 : bi);
        const float rd = xr - (float)ri;
        const float gd = xg - (float)gi;
        const float bd = xb - (float)bi;

        const int base = (bi * DIM + gi) * DIM + ri;
        const int o000 = base,            o001 = base + 1;
        const int o010 = base + DIM,      o011 = base + DIM + 1;
        const int o100 = base + DD,       o101 = base + DD + 1;
        const int o110 = base + DD + DIM, o111 = base + DD + DIM + 1;

        const float w000 = (1.f - rd) * (1.f - gd) * (1.f - bd);
        const float w001 = rd * (1.f - gd) * (1.f - bd);
        const float w010 = (1.f - rd) * gd * (1.f - bd);
        const float w011 = rd * gd * (1.f - bd);
        const float w100 = (1.f - rd) * (1.f - gd) * bd;
        const float w101 = rd * (1.f - gd) * bd;
        const float w110 = (1.f - rd) * gd * bd;
        const float w111 = rd * gd * bd;

        o0p[j] = r  + L0[o000] * w000 + L0[o001] * w001 + L0[o010] * w010 + L0[o011] * w011
                    + L0[o100] * w100 + L0[o101] * w101 + L0[o110] * w110 + L0[o111] * w111;
        o1p[j] = g  + L1[o000] * w000 + L1[o001] * w001 + L1[o010] * w010 + L1[o011] * w011
                    + L1[o100] * w100 + L1[o101] * w101 + L1[o110] * w110 + L1[o111] * w111;
        o2p[j] = bl + L2[o000] * w000 + L2[o001] * w001 + L2[o010] * w010 + L2[o011] * w011
                    + L2[o100] * w100 + L2[o101] * w101 + L2[o110] * w110 + L2[o111] * w111;
    }

    *(float4*)(out + ib)                  = o0;
    *(float4*)(out + ib + HW)             = o1;
    *(float4*)(out + ib + 2 * (size_t)HW) = o2;
}

// ---------------------------------------------------------------------------
extern "C" void kernel_launch(void* const* d_in, const int* in_sizes, int n_in,
                              void* d_out, int out_size, void* d_ws, size_t ws_size,
                              hipStream_t stream) {
    (void)in_sizes; (void)n_in; (void)out_size; (void)ws_size;

    const float* img  = (const float*)d_in[0];   // (8,3,1024,1024)
    const float* pw   = (const float*)d_in[1];   // (8,20)
    const float* sl   = (const float*)d_in[2];   // (33,5)
    const float* wlay = (const float*)d_in[3];   // (20,1089)
    const float* luts = (const float*)d_in[4];   // (300,20)
    float*       out  = (float*)d_out;

    float* ws  = (float*)d_ws;
    float* T   = ws;                 // 326700 floats (300x1089)
    float* V   = ws + 326784;        // 130680 floats (8*3*5*1089)
    float* LUT = ws + 457856;        // 862488 floats (8*3*33^3)

    // Stage 1: WMMA f32 GEMM, 19x69 tiles, 1 wave each
    dim3 g1(19, 69);
    nlut_gemm_wmma<<<g1, 32, 0, stream>>>(luts, wlay, T);

    // Stage 2: blend with pred_weight
    const int t2 = NBAT * 3 * SL * DD;
    nlut_fuse_pw<<<(t2 + 255) / 256, 256, 0, stream>>>(T, pw, V);

    // Stage 3: s_layers contraction + cube_to_lut permutation
    const int t3 = NBAT * 3 * DDD;
    nlut_build_lut<<<(t3 + 255) / 256, 256, 0, stream>>>(V, sl, LUT);

    // Stage 4: trilinear apply + residual (memory bound)
    const int quads = NBAT * (HW / 4);           // 2,097,152
    nlut_apply<<<quads / 256, 256, 0, stream>>>(img, LUT, out);
}